// VGNet_Feature_84129819394148
// MI455X (gfx1250) — compile-verified
//
#include <hip/hip_runtime.h>

typedef __attribute__((ext_vector_type(2))) float v2f;
typedef __attribute__((ext_vector_type(8))) float v8f;

#define NPTS 32768
#define NB 8
#define NEG_SLOPE 0.01f
#define TILES_PER_B (NPTS / 16)         // 2048
#define TOTAL_TILES (NB * TILES_PER_B)  // 16384

// ---------- WMMA helpers (fp32 16x16x4) ----------

__device__ __forceinline__ v8f wmma4(v2f a, v2f b, v8f c) {
  // D = A(16x4) * B(4x16) + C, full fp32
  return __builtin_amdgcn_wmma_f32_16x16x4_f32(false, a, false, b, (short)0, c,
                                               false, false);
}

// leaky_relu(x) == max(x, slope*x) for 0 < slope < 1: single v_max_f32 per elem
__device__ __forceinline__ v8f lrelu8(v8f v) {
#pragma unroll
  for (int i = 0; i < 8; ++i) v[i] = fmaxf(v[i], NEG_SLOPE * v[i]);
  return v;
}

// ---------- monotonic float-max keys (order-independent atomic max) ----------

__device__ __forceinline__ unsigned f2key(float f) {
  unsigned u = __float_as_uint(f);
  return (u & 0x80000000u) ? ~u : (u | 0x80000000u);
}
__device__ __forceinline__ float key2f(unsigned k) {
  unsigned u = (k & 0x80000000u) ? (k & 0x7fffffffu) : ~k;
  return __uint_as_float(u);
}

// ---------- conv building blocks ----------
// Activation layout in LDS: act[chan * 16 + col], 16 points per wave tile.
// A-frag (16x4): lane l -> row = l&15, K = k0 + (l>=16 ? 2:0) .. +1
// B-frag (4x16): lane l -> col = l&15, K = k0 + (l>=16 ? 2:0) .. +1
// C/D      : lane l, vgpr r -> row = r + (l>=16 ? 8:0), col = l&15

// first layer: Cin=3 (padded to K=4), Cout=16, input from global [pt][3]
__device__ void conv3(const float* __restrict__ W, const float* __restrict__ Bv,
                      const float* __restrict__ gin, float* actOut, int lane) {
  const int col = lane & 15;
  const int hi = lane >> 4;          // 0 or 1
  const int kh = hi * 2;             // 0 or 2
  const int rsel = hi * 8;           // 0 or 8
  v8f acc;
#pragma unroll
  for (int r = 0; r < 8; ++r) acc[r] = Bv[r + rsel];
  v2f a, bb;
  // kh==0 -> (c0,c1); kh==2 -> (c2, 0). Load from always-valid address, then
  // select — avoids an exec-mask branch around the second load.
  a.x = W[col * 3 + kh];
  float wy = W[col * 3 + (hi ? 2 : 1)];
  a.y = hi ? 0.f : wy;
  bb.x = gin[col * 3 + kh];
  float gy = gin[col * 3 + (hi ? 2 : 1)];
  bb.y = hi ? 0.f : gy;
  acc = wmma4(a, bb, acc);
  acc = lrelu8(acc);
#pragma unroll
  for (int r = 0; r < 8; ++r) actOut[(r + rsel) * 16 + col] = acc[r];
}

// generic conv from LDS activations, CIN % 4 == 0, COUT % 16 == 0.
// K-outer / M-inner: COUT/16 independent WMMA chains in flight, B fragment
// loaded once per K-step and reused across all output tiles.
template <int CIN, int COUT>
__device__ void conv_lds(const float* __restrict__ W, const float* __restrict__ Bv,
                         const float* actIn, float* actOut, int lane) {
  constexpr int MT = COUT / 16;
  const int col = lane & 15;
  const int kh = (lane >> 4) * 2;
  const int rsel = (lane >> 4) * 8;
  v8f acc[MT];
#pragma unroll
  for (int mt = 0; mt < MT; ++mt)
#pragma unroll
    for (int r = 0; r < 8; ++r) acc[mt][r] = Bv[mt * 16 + r + rsel];
  for (int k0 = 0; k0 < CIN; k0 += 4) {
    const int kb = k0 + kh;
    v2f bb;
    bb.x = actIn[kb * 16 + col];
    bb.y = actIn[(kb + 1) * 16 + col];
#pragma unroll
    for (int mt = 0; mt < MT; ++mt) {
      v2f a;
      a.x = W[(mt * 16 + col) * CIN + kb];
      a.y = W[(mt * 16 + col) * CIN + kb + 1];
      acc[mt] = wmma4(a, bb, acc[mt]);
    }
  }
#pragma unroll
  for (int mt = 0; mt < MT; ++mt) {
    v8f d = lrelu8(acc[mt]);
#pragma unroll
    for (int r = 0; r < 8; ++r) actOut[(mt * 16 + r + rsel) * 16 + col] = d[r];
  }
}

// concat input: k<112 -> bufA (per-point feats), 112..223 -> gf broadcast, 224.. -> bufN
__device__ __forceinline__ float readConcat(const float* bufA, const float* gf,
                                            const float* bufN, int k, int col) {
  if (k < 112) return bufA[k * 16 + col];
  if (k < 224) return gf[k - 112];
  return bufN[(k - 224) * 16 + col];
}

// conv21: 336 -> 64 on the concat features (4 independent accumulator chains)
__device__ void conv336(const float* __restrict__ W, const float* __restrict__ Bv,
                        const float* bufA, const float* gf, const float* bufN,
                        float* actOut, int lane) {
  const int col = lane & 15;
  const int kh = (lane >> 4) * 2;
  const int rsel = (lane >> 4) * 8;
  v8f acc[4];
#pragma unroll
  for (int mt = 0; mt < 4; ++mt)
#pragma unroll
    for (int r = 0; r < 8; ++r) acc[mt][r] = Bv[mt * 16 + r + rsel];
  for (int k0 = 0; k0 < 336; k0 += 4) {
    const int kb = k0 + kh;
    v2f bb;
    bb.x = readConcat(bufA, gf, bufN, kb, col);
    bb.y = readConcat(bufA, gf, bufN, kb + 1, col);
#pragma unroll
    for (int mt = 0; mt < 4; ++mt) {
      v2f a;
      a.x = W[(mt * 16 + col) * 336 + kb];
      a.y = W[(mt * 16 + col) * 336 + kb + 1];
      acc[mt] = wmma4(a, bb, acc[mt]);
    }
  }
#pragma unroll
  for (int mt = 0; mt < 4; ++mt) {
    v8f d = lrelu8(acc[mt]);
#pragma unroll
    for (int r = 0; r < 8; ++r) actOut[(mt * 16 + r + rsel) * 16 + col] = d[r];
  }
}

// max over 16 points for each of 112 channels -> global atomic max keys
__device__ void maxreduce112(const float* buf, unsigned* __restrict__ keys, int lane) {
  for (int c = lane; c < 112; c += 32) {
    float m = buf[c * 16];
#pragma unroll
    for (int j = 1; j < 16; ++j) m = fmaxf(m, buf[c * 16 + j]);
    atomicMax(&keys[c], f2key(m));
  }
}

// ---------- parameters ----------

struct NetParams {
  const float* x;
  const float* r;
  const float* nrm;
  const float* w[15];
  const float* b[15];
  unsigned* gf1k;   // [NB*112]
  unsigned* rgf1k;  // [NB*112]
  unsigned* gf2k;   // [NB*112]
  unsigned* rgf2k;  // [NB*112]
};
// layer indices: 0..2 xyz11-13, 3..5 xyz21-23, 6..8 rgb11-13, 9..11 rgb21-23,
//                12..14 normal11-13

// ---------- kernels ----------

__global__ void initws_kernel(unsigned* ws) {
  int i = blockIdx.x * blockDim.x + threadIdx.x;
  if (i < 4 * NB * 112) ws[i] = 0u;  // key 0 < key(any float)
}

// phase 1: chain1 for xyz and rgb, global max -> gf1 / rgb_gf1 keys
__global__ void __launch_bounds__(256) phase1_kernel(NetParams P) {
  __shared__ float buf[8][112 * 16];
  const int wave = threadIdx.x >> 5;
  const int lane = threadIdx.x & 31;
  const int t = blockIdx.x * 8 + wave;
  const int b = t / TILES_PER_B;
  const int p0 = (t % TILES_PER_B) * 16;
  float* bf = buf[wave];
  const float* xin = P.x + ((size_t)b * NPTS + p0) * 3;
  const float* rin = P.r + ((size_t)b * NPTS + p0) * 3;

  // xyz chain1: 3->16->32->64, concat channels [h1|h2|h3] = 0..111
  conv3(P.w[0], P.b[0], xin, bf, lane);
  __syncthreads();
  conv_lds<16, 32>(P.w[1], P.b[1], bf, bf + 16 * 16, lane);
  __syncthreads();
  conv_lds<32, 64>(P.w[2], P.b[2], bf + 16 * 16, bf + 48 * 16, lane);
  __syncthreads();
  maxreduce112(bf, P.gf1k + b * 112, lane);
  __syncthreads();

  // rgb chain1 reusing the same buffer
  conv3(P.w[6], P.b[6], rin, bf, lane);
  __syncthreads();
  conv_lds<16, 32>(P.w[7], P.b[7], bf, bf + 16 * 16, lane);
  __syncthreads();
  conv_lds<32, 64>(P.w[8], P.b[8], bf + 16 * 16, bf + 48 * 16, lane);
  __syncthreads();
  maxreduce112(bf, P.rgf1k + b * 112, lane);
}

// phase 2: recompute chain1 feats, run chain2 with gf broadcast, max -> gf2 keys
__global__ void __launch_bounds__(64) phase2_kernel(NetParams P) {
  __shared__ float buf1[2][112 * 16];  // xyz/rgb per-point chain1 feats
  __shared__ float bufN[2][112 * 16];  // normal chain1 feats
  __shared__ float buf2[2][112 * 16];  // chain2 feats
  __shared__ float gfs[112];
  __shared__ float rgfs[112];
  const int wave = threadIdx.x >> 5;
  const int lane = threadIdx.x & 31;
  const int t = blockIdx.x * 2 + wave;
  const int b = t / TILES_PER_B;
  const int p0 = (t % TILES_PER_B) * 16;
  float* b1 = buf1[wave];
  float* bN = bufN[wave];
  float* b2 = buf2[wave];
  const float* xin = P.x + ((size_t)b * NPTS + p0) * 3;
  const float* rin = P.r + ((size_t)b * NPTS + p0) * 3;
  const float* nin = P.nrm + ((size_t)b * NPTS + p0) * 3;

  for (int c = threadIdx.x; c < 112; c += 64) {
    gfs[c] = key2f(P.gf1k[b * 112 + c]);
    rgfs[c] = key2f(P.rgf1k[b * 112 + c]);
  }
  __syncthreads();

  // normal chain1 -> bufN
  conv3(P.w[12], P.b[12], nin, bN, lane);
  __syncthreads();
  conv_lds<16, 32>(P.w[13], P.b[13], bN, bN + 16 * 16, lane);
  __syncthreads();
  conv_lds<32, 64>(P.w[14], P.b[14], bN + 16 * 16, bN + 48 * 16, lane);
  __syncthreads();

  // xyz chain1 -> buf1
  conv3(P.w[0], P.b[0], xin, b1, lane);
  __syncthreads();
  conv_lds<16, 32>(P.w[1], P.b[1], b1, b1 + 16 * 16, lane);
  __syncthreads();
  conv_lds<32, 64>(P.w[2], P.b[2], b1 + 16 * 16, b1 + 48 * 16, lane);
  __syncthreads();

  // xyz chain2: concat(112 pf | 112 gf | 112 normal) -> 64 -> 32 -> 16
  conv336(P.w[3], P.b[3], b1, gfs, bN, b2, lane);
  __syncthreads();
  conv_lds<64, 32>(P.w[4], P.b[4], b2, b2 + 64 * 16, lane);
  __syncthreads();
  conv_lds<32, 16>(P.w[5], P.b[5], b2 + 64 * 16, b2 + 96 * 16, lane);
  __syncthreads();
  maxreduce112(b2, P.gf2k + b * 112, lane);
  __syncthreads();

  // rgb chain1 -> buf1 (reuse)
  conv3(P.w[6], P.b[6], rin, b1, lane);
  __syncthreads();
  conv_lds<16, 32>(P.w[7], P.b[7], b1, b1 + 16 * 16, lane);
  __syncthreads();
  conv_lds<32, 64>(P.w[8], P.b[8], b1 + 16 * 16, b1 + 48 * 16, lane);
  __syncthreads();

  // rgb chain2
  conv336(P.w[9], P.b[9], b1, rgfs, bN, b2, lane);
  __syncthreads();
  conv_lds<64, 32>(P.w[10], P.b[10], b2, b2 + 64 * 16, lane);
  __syncthreads();
  conv_lds<32, 16>(P.w[11], P.b[11], b2 + 64 * 16, b2 + 96 * 16, lane);
  __syncthreads();
  maxreduce112(b2, P.rgf2k + b * 112, lane);
}

__global__ void finalize_kernel(const unsigned* __restrict__ gf2k,
                                const unsigned* __restrict__ rgf2k,
                                float* __restrict__ out) {
  int i = blockIdx.x * blockDim.x + threadIdx.x;
  if (i >= NB * 224) return;
  int b = i / 224, c = i % 224;
  out[i] = (c < 112) ? key2f(gf2k[b * 112 + c]) : key2f(rgf2k[b * 112 + c - 112]);
}

// ---------- launch ----------

extern "C" void kernel_launch(void* const* d_in, const int* in_sizes, int n_in,
                              void* d_out, int out_size, void* d_ws, size_t ws_size,
                              hipStream_t stream) {
  NetParams P;
  P.x = (const float*)d_in[0];
  P.r = (const float*)d_in[1];
  P.nrm = (const float*)d_in[2];
  for (int i = 0; i < 15; ++i) {
    P.w[i] = (const float*)d_in[3 + 2 * i];
    P.b[i] = (const float*)d_in[4 + 2 * i];
  }
  unsigned* ws = (unsigned*)d_ws;
  P.gf1k = ws;
  P.rgf1k = ws + NB * 112;
  P.gf2k = ws + 2 * NB * 112;
  P.rgf2k = ws + 3 * NB * 112;

  initws_kernel<<<(4 * NB * 112 + 255) / 256, 256, 0, stream>>>(ws);
  phase1_kernel<<<TOTAL_TILES / 8, 256, 0, stream>>>(P);
  phase2_kernel<<<TOTAL_TILES / 2, 64, 0, stream>>>(P);
  finalize_kernel<<<(NB * 224 + 255) / 256, 256, 0, stream>>>(P.gf2k, P.rgf2k,
                                                              (float*)d_out);
}